// NnHalfKA_13580686590393
// MI455X (gfx1250) — compile-verified
//
#include <hip/hip_runtime.h>
#include <math.h>

#define B_SZ   16384
#define NNZ_PER 32
#define NNZ    (B_SZ * NNZ_PER)
#define FT_OUT 512
#define F_BIG  49152
#define F_SMALL 768
#define MODV   640

typedef float v2f __attribute__((ext_vector_type(2)));
typedef float v8f __attribute__((ext_vector_type(8)));

// Static device-side scratch (no dependence on ws_size; rewritten every call).
__device__ float g_WftT[(size_t)F_BIG * FT_OUT];     // 96 MB  -> L2-resident gather table
__device__ float g_WfftT[(size_t)F_SMALL * FT_OUT];  // 1.5 MB
__device__ float g_fft_acc[2][(size_t)MODV * FT_OUT];// 2.5 MB

// ---------------------------------------------------------------------------
// LDS-tiled transpose: dst[c*512 + r] = src[r*cols + c]   (rows fixed = 512)
// which==0 -> g_WftT (cols=49152), which==1 -> g_WfftT (cols=768)
// ---------------------------------------------------------------------------
__global__ __launch_bounds__(256) void transpose_k(const float* __restrict__ src,
                                                   int cols, int which) {
  float* __restrict__ dst = (which == 0) ? g_WftT : g_WfftT;
  __shared__ float tile[32][33];
  const int c0 = blockIdx.x * 32;
  const int r0 = blockIdx.y * 32;
  const int tx = threadIdx.x, ty = threadIdx.y;  // block (32, 8)
#pragma unroll
  for (int i = 0; i < 32; i += 8)
    tile[ty + i][tx] = src[(size_t)(r0 + ty + i) * cols + (c0 + tx)];
  __syncthreads();
#pragma unroll
  for (int i = 0; i < 32; i += 8)
    dst[(size_t)(c0 + ty + i) * FT_OUT + (r0 + tx)] = tile[tx][ty + i];
}

// ---------------------------------------------------------------------------
// FFT (%640) pass via WMMA. One wave per (side, out-row r, 16-col tile).
// D = ones(16x4) x B(4x16) accumulated over all nnz with pos%640 == r.
// A == all-ones makes the result invariant to the K<->(vgpr,lane-half)
// mapping, so the B-operand layout assumption cannot corrupt the sum.
// ---------------------------------------------------------------------------
__global__ __launch_bounds__(256) void fft_accum_k(const long long* __restrict__ stm_cols,
                                                   const long long* __restrict__ nstm_cols,
                                                   const float* __restrict__ values) {
  const int wid  = blockIdx.x * 8 + (threadIdx.x >> 5);   // 8 waves / block
  const int lane = threadIdx.x & 31;
  const int side = wid / (MODV * 32);
  const int rem  = wid % (MODV * 32);
  const int r    = rem >> 5;
  const int tile = rem & 31;
  const int n    = lane & 15;        // column within tile (N index of B/D)
  const int h    = lane >> 4;        // lane-half selects K rows {0,1} vs {2,3}
  const int col  = tile * 16 + n;

  const long long* __restrict__ cols = side ? nstm_cols : stm_cols;

  v8f acc = {0.f, 0.f, 0.f, 0.f, 0.f, 0.f, 0.f, 0.f};
  v2f a   = {1.0f, 1.0f};            // all-ones A (layout-independent)

  for (int p = r; p < B_SZ; p += MODV) {
    const long long* __restrict__ cp = cols + (size_t)p * NNZ_PER;
    const float* __restrict__ vp = values + (size_t)p * NNZ_PER;
#pragma unroll
    for (int j = 0; j < NNZ_PER; j += 4) {
      const int k0 = j + 2 * h;      // B vgpr0 holds K = j+0 / j+2 per half
      const int k1 = k0 + 1;         // B vgpr1 holds K = j+1 / j+3 per half
      const int c0 = (int)(cp[k0] % MODV);
      const int c1 = (int)(cp[k1] % MODV);
      v2f b;
      b.x = g_WfftT[(size_t)c0 * FT_OUT + col] * vp[k0];
      b.y = g_WfftT[(size_t)c1 * FT_OUT + col] * vp[k1];
      acc = __builtin_amdgcn_wmma_f32_16x16x4_f32(
          /*neg_a=*/false, a, /*neg_b=*/false, b,
          /*c_mod=*/(short)0, acc, /*reuse_a=*/false, /*reuse_b=*/false);
    }
  }
  // All D rows are identical (identical A rows); lane l holds column l&15.
  if (h == 0)
    g_fft_acc[side][(size_t)r * FT_OUT + col] = acc[0];
}

// ---------------------------------------------------------------------------
// Main fused kernel: one 128-thread block per position.
// Gather-accumulate 32 rows x 2 sides from L2-resident g_WftT (float4/lane),
// add biases + fft contribution (p < 640), clip, output dot, sigmoid.
// ---------------------------------------------------------------------------
__global__ __launch_bounds__(128) void ft_main_k(const long long* __restrict__ stm_cols,
                                                 const long long* __restrict__ nstm_cols,
                                                 const float* __restrict__ values,
                                                 const float* __restrict__ b_ft,
                                                 const float* __restrict__ b_fft,
                                                 const float* __restrict__ W_out,
                                                 const float* __restrict__ b_out,
                                                 float* __restrict__ out) {
  const int p = blockIdx.x;
  const int t = threadIdx.x;

  __shared__ int   cs[NNZ_PER];
  __shared__ int   cn[NNZ_PER];
  __shared__ float vv[NNZ_PER];
  __shared__ float red[128];

  if (t < NNZ_PER) {
    cs[t] = (int)stm_cols[(size_t)p * NNZ_PER + t];
    cn[t] = (int)nstm_cols[(size_t)p * NNZ_PER + t];
    vv[t] = values[(size_t)p * NNZ_PER + t];
  }
  __syncthreads();

  const int off = t * 4;  // 128 threads x float4 = 512 features
  const float4 bf  = *(const float4*)(b_ft + off);
  const float4 bff = *(const float4*)(b_fft + off);

  float accs[4] = {bf.x + bff.x, bf.y + bff.y, bf.z + bff.z, bf.w + bff.w};
  float accn[4] = {accs[0], accs[1], accs[2], accs[3]};

  if (p < MODV) {  // segment_sum only populates rows 0..639 in the %640 pass
    const float4 fs = *(const float4*)&g_fft_acc[0][(size_t)p * FT_OUT + off];
    const float4 fn = *(const float4*)&g_fft_acc[1][(size_t)p * FT_OUT + off];
    accs[0] += fs.x; accs[1] += fs.y; accs[2] += fs.z; accs[3] += fs.w;
    accn[0] += fn.x; accn[1] += fn.y; accn[2] += fn.z; accn[3] += fn.w;
  }

#pragma unroll 4
  for (int j = 0; j < NNZ_PER; ++j) {
    if (j + 2 < NNZ_PER) {  // global_prefetch_b8 the upcoming gather rows
      __builtin_prefetch(&g_WftT[(size_t)cs[j + 2] * FT_OUT + off], 0, 1);
      __builtin_prefetch(&g_WftT[(size_t)cn[j + 2] * FT_OUT + off], 0, 1);
    }
    const float4 rs = *(const float4*)&g_WftT[(size_t)cs[j] * FT_OUT + off];
    const float4 rn = *(const float4*)&g_WftT[(size_t)cn[j] * FT_OUT + off];
    const float v = vv[j];
    accs[0] = fmaf(v, rs.x, accs[0]); accs[1] = fmaf(v, rs.y, accs[1]);
    accs[2] = fmaf(v, rs.z, accs[2]); accs[3] = fmaf(v, rs.w, accs[3]);
    accn[0] = fmaf(v, rn.x, accn[0]); accn[1] = fmaf(v, rn.y, accn[1]);
    accn[2] = fmaf(v, rn.z, accn[2]); accn[3] = fmaf(v, rn.w, accn[3]);
  }

  // clip(hidden, 0, 1), then partial dot with W_out ([0:512]=stm, [512:1024]=nstm)
  const float4 ws = *(const float4*)(W_out + off);
  const float4 wn = *(const float4*)(W_out + FT_OUT + off);
  float partial = 0.f;
  partial = fmaf(fminf(fmaxf(accs[0], 0.f), 1.f), ws.x, partial);
  partial = fmaf(fminf(fmaxf(accs[1], 0.f), 1.f), ws.y, partial);
  partial = fmaf(fminf(fmaxf(accs[2], 0.f), 1.f), ws.z, partial);
  partial = fmaf(fminf(fmaxf(accs[3], 0.f), 1.f), ws.w, partial);
  partial = fmaf(fminf(fmaxf(accn[0], 0.f), 1.f), wn.x, partial);
  partial = fmaf(fminf(fmaxf(accn[1], 0.f), 1.f), wn.y, partial);
  partial = fmaf(fminf(fmaxf(accn[2], 0.f), 1.f), wn.z, partial);
  partial = fmaf(fminf(fmaxf(accn[3], 0.f), 1.f), wn.w, partial);

  red[t] = partial;
  __syncthreads();
  for (int s = 64; s > 0; s >>= 1) {
    if (t < s) red[t] += red[t + s];
    __syncthreads();
  }
  if (t == 0)
    out[p] = 1.0f / (1.0f + expf(-(red[0] + b_out[0])));
}

// ---------------------------------------------------------------------------
extern "C" void kernel_launch(void* const* d_in, const int* in_sizes, int n_in,
                              void* d_out, int out_size, void* d_ws, size_t ws_size,
                              hipStream_t stream) {
  const long long* stm    = (const long long*)d_in[0];  // [2, NNZ] int64
  const long long* nstm   = (const long long*)d_in[1];
  const float*     values = (const float*)d_in[2];
  // d_in[3] = size (unused; B is compile-time constant)
  const float*     W_ft   = (const float*)d_in[4];      // [512, 49152]
  const float*     b_ft   = (const float*)d_in[5];
  const float*     W_fft  = (const float*)d_in[6];      // [512, 768]
  const float*     b_fft  = (const float*)d_in[7];
  const float*     W_out  = (const float*)d_in[8];      // [1, 1024]
  const float*     b_out  = (const float*)d_in[9];
  float*           out    = (float*)d_out;

  const long long* stm_cols  = stm  + NNZ;  // row 1 of the [2, NNZ] stack
  const long long* nstm_cols = nstm + NNZ;

  // 1) One-time-per-call transposes (streaming; ~192 MB @ 23.3 TB/s)
  transpose_k<<<dim3(F_BIG / 32, FT_OUT / 32), dim3(32, 8), 0, stream>>>(W_ft, F_BIG, 0);
  transpose_k<<<dim3(F_SMALL / 32, FT_OUT / 32), dim3(32, 8), 0, stream>>>(W_fft, F_SMALL, 1);

  // 2) %640 FFT accumulation on the matrix pipe: 2*640*32 waves, 8 waves/block
  fft_accum_k<<<(2 * MODV * 32) / 8, 256, 0, stream>>>(stm_cols, nstm_cols, values);

  // 3) Fused gather + biases + clip + output layer + sigmoid
  ft_main_k<<<B_SZ, 128, 0, stream>>>(stm_cols, nstm_cols, values,
                                      b_ft, b_fft, W_out, b_out, out);
}